// KineticForecastingFramework_33706903339137
// MI455X (gfx1250) — compile-verified
//
#include <hip/hip_runtime.h>
#include <math.h>

#define N_NODES 50000
#define E_EDGES 800000
#define QDIM 64
#define NLAYERS 6
#define DT_STEP 0.1f
#define XI_MAX 70.0f
#define NTILES (N_NODES / 16)   // 3125 exactly

typedef float v2f __attribute__((ext_vector_type(2)));
typedef float v8f __attribute__((ext_vector_type(8)));

// Non-returning hardware fp32 atomic add (GLOBAL_ATOMIC_ADD_F32, device scope).
__device__ __forceinline__ void atomic_add_f32(float* p, float v) {
    __hip_atomic_fetch_add(p, v, __ATOMIC_RELAXED, __HIP_MEMORY_SCOPE_AGENT);
}

// ---------------------------------------------------------------------------
// K1: g = relu(f); acc = 0; deg = 0
// ---------------------------------------------------------------------------
__global__ void k_prep(const float* __restrict__ f, float* __restrict__ g,
                       float* __restrict__ acc, float* __restrict__ deg) {
    int i = blockIdx.x * blockDim.x + threadIdx.x;
    if (i < N_NODES * QDIM) {
        float v = f[i];
        g[i]   = v > 0.f ? v : 0.f;
        acc[i] = 0.f;
    }
    if (i < N_NODES) deg[i] = 0.f;
}

// ---------------------------------------------------------------------------
// K2: out-degree of src
// ---------------------------------------------------------------------------
__global__ void k_deg(const int* __restrict__ src, float* __restrict__ deg) {
    int e = blockIdx.x * blockDim.x + threadIdx.x;
    if (e < E_EDGES) atomic_add_f32(&deg[src[e]], 1.0f);
}

// ---------------------------------------------------------------------------
// K3: per-edge transport scatter. One wave (32 lanes) per edge; q = lane,
// lane+32 (coalesced 128B rows). transport = outflow - inflow folds into a
// single accumulator: acc[src] += msg, acc[dst] -= msg. Edge metadata is
// wave-uniform -> pin it to SGPRs via readfirstlane.
// ---------------------------------------------------------------------------
__global__ void __launch_bounds__(256)
k_transport(const float* __restrict__ g, const float* __restrict__ w,
            const int* __restrict__ src, const int* __restrict__ dst,
            const float* __restrict__ deg, float* __restrict__ acc) {
    int e    = blockIdx.x * 8 + (threadIdx.x >> 5);
    int lane = threadIdx.x & 31;
    if (e >= E_EDGES) return;

    // uniform per-wave scalars
    int s = __builtin_amdgcn_readfirstlane(src[e]);
    int d = __builtin_amdgcn_readfirstlane(dst[e]);
    float dg    = deg[s];
    float coef0 = (dg > 0.f ? 1.0f / dg : 0.0f) * w[e];
    coef0 = __builtin_amdgcn_readfirstlane(coef0);

    const float xi_step = XI_MAX / (float)(QDIM - 1);
    const float* gs = g + (size_t)s * QDIM;
    const float* gd = g + (size_t)d * QDIM;
    float* as = acc + (size_t)s * QDIM;
    float* ad = acc + (size_t)d * QDIM;
#pragma unroll
    for (int it = 0; it < 2; ++it) {
        int   q  = lane + it * 32;
        float xi = xi_step * (float)q;
        float m  = coef0 * xi * (gd[q] - gs[q]);
        atomic_add_f32(&as[q],  m);
        atomic_add_f32(&ad[q], -m);
    }
}

// ---------------------------------------------------------------------------
// K4: collision MLP (6x 64x64 fp32 GEMM via V_WMMA_F32_16X16X4_F32) fused
// with the final update. One wave per 16-node tile, 8 tiles per block.
//
// WMMA fragment layouts (ISA 7.12.2, wave32):
//   A (16x4, M x K): lane -> M = lane%16; VGPR{0,1} hold K = 2*(lane/16)+{0,1}
//   B (4x16, K x N): lane -> N = lane%16; VGPR{0,1} hold K = 2*(lane/16)+{0,1}
//   C/D (16x16):     lane -> N = lane%16; VGPR v -> M = v + 8*(lane/16)
// So with activations staged in LDS as [m][k] and W rows as [h][k], both the
// A and B fragments are single 8-byte LDS loads (compiler pairs them into
// ds_load_2addr_b64).
// ---------------------------------------------------------------------------
__global__ void __launch_bounds__(256)
k_collision_update(const float* __restrict__ g, const float* __restrict__ acc,
                   const float* __restrict__ srcterm,
                   const float* __restrict__ mlpW, const float* __restrict__ mlpB,
                   float* __restrict__ out) {
    __shared__ float sW[QDIM * QDIM];        // 16 KB, one layer at a time
    __shared__ float sB[QDIM];
    __shared__ float stage[8][16 * QDIM];    // 4 KB per wave: activations [m][k]

    const int tid  = threadIdx.x;
    const int wave = tid >> 5;
    const int lane = tid & 31;
    const int half = lane >> 4;              // which 16-lane half
    const int l16  = lane & 15;

    const int  tile  = blockIdx.x * 8 + wave;
    const bool valid = (tile < NTILES);
    const int  node0 = tile * 16;

    float* st = stage[wave];

    // Load this wave's 16x64 relu(f) tile into LDS staging (coalesced f4).
    if (valid) {
        const float4* gs = (const float4*)(g + (size_t)node0 * QDIM);
        float4*       sd = (float4*)st;
#pragma unroll
        for (int i = 0; i < 8; ++i) sd[lane + i * 32] = gs[lane + i * 32];
    }

    for (int layer = 0; layer < NLAYERS; ++layer) {
        __syncthreads();
        // Stage weights + bias of this layer (4096 floats, 256 threads).
        {
            const float4* ws = (const float4*)(mlpW + (size_t)layer * QDIM * QDIM);
            float4*       wd = (float4*)sW;
#pragma unroll
            for (int i = 0; i < 4; ++i) wd[tid + i * 256] = ws[tid + i * 256];
            if (tid < QDIM) sB[tid] = mlpB[layer * QDIM + tid];
        }
        __syncthreads();

        if (valid) {
            // 4 output h-tiles, C initialized with the bias (C[m][h] = b[h]).
            v8f ct[4];
#pragma unroll
            for (int ht = 0; ht < 4; ++ht) {
                float b = sB[ht * 16 + l16];
#pragma unroll
                for (int j = 0; j < 8; ++j) ct[ht][j] = b;
            }

            // K = 64 in 16 steps of 4.
#pragma unroll
            for (int gk = 0; gk < 16; ++gk) {
                int kbase = gk * 4 + half * 2;
                v2f afrag = *(const v2f*)&st[l16 * QDIM + kbase];   // x[m][k..k+1]
#pragma unroll
                for (int ht = 0; ht < 4; ++ht) {
                    int h = ht * 16 + l16;
                    v2f bfrag = *(const v2f*)&sW[h * QDIM + kbase]; // W[h][k..k+1] = B^T
                    ct[ht] = __builtin_amdgcn_wmma_f32_16x16x4_f32(
                        false, afrag, false, bfrag, (short)0, ct[ht], false, false);
                }
            }

            if (layer < NLAYERS - 1) {
                // relu, write back to staging in [m][k] layout for next layer.
#pragma unroll
                for (int ht = 0; ht < 4; ++ht)
#pragma unroll
                    for (int v = 0; v < 8; ++v) {
                        float x = ct[ht][v];
                        st[(v + 8 * half) * QDIM + ht * 16 + l16] = x > 0.f ? x : 0.f;
                    }
            } else {
                // tanh + fused final update: relu(g - DT*(transport - coll - src))
#pragma unroll
                for (int ht = 0; ht < 4; ++ht)
#pragma unroll
                    for (int v = 0; v < 8; ++v) {
                        float coll = tanhf(ct[ht][v]);
                        int   m    = v + 8 * half;
                        int   h    = ht * 16 + l16;
                        size_t idx = (size_t)(node0 + m) * QDIM + h;
                        float fn = g[idx] - DT_STEP * (acc[idx] - coll - srcterm[idx]);
                        out[idx] = fn > 0.f ? fn : 0.f;
                    }
            }
        }
    }
}

// ---------------------------------------------------------------------------
extern "C" void kernel_launch(void* const* d_in, const int* in_sizes, int n_in,
                              void* d_out, int out_size, void* d_ws, size_t ws_size,
                              hipStream_t stream) {
    const float* f       = (const float*)d_in[0];  // [N,Q]
    const float* w       = (const float*)d_in[1];  // [E]
    const float* srcterm = (const float*)d_in[2];  // [N,Q]
    const float* mlpW    = (const float*)d_in[3];  // [NL,Q,Q]
    const float* mlpB    = (const float*)d_in[4];  // [NL,Q]
    const int*   src     = (const int*)d_in[5];    // [E]
    const int*   dst     = (const int*)d_in[6];    // [E]
    float*       out     = (float*)d_out;

    float* g   = (float*)d_ws;                  // relu(f)      N*Q
    float* acc = g + (size_t)N_NODES * QDIM;    // transport    N*Q
    float* deg = acc + (size_t)N_NODES * QDIM;  // out-degree   N

    k_prep<<<(N_NODES * QDIM + 255) / 256, 256, 0, stream>>>(f, g, acc, deg);
    k_deg<<<(E_EDGES + 255) / 256, 256, 0, stream>>>(src, deg);
    k_transport<<<(E_EDGES + 7) / 8, 256, 0, stream>>>(g, w, src, dst, deg, acc);
    k_collision_update<<<(NTILES + 7) / 8, 256, 0, stream>>>(g, acc, srcterm, mlpW, mlpB, out);
}